// MGMIN_FSA_51299089383490
// MI455X (gfx1250) — compile-verified
//
#include <hip/hip_runtime.h>
#include <math.h>

typedef __attribute__((ext_vector_type(2))) float v2f;
typedef __attribute__((ext_vector_type(8))) float v8f;

#define S_DIM 192
#define H_DIM 128
#define KD    50
#define NB    8
#define KPHI  (H_DIM * 9)   // 1152 expanded features: 128 silu + 128*8 spline

__device__ __forceinline__ float silu_f(float x) {
    return x / (1.0f + __expf(-x));
}

// Cox-de Boor cubic B-spline basis, uniform knots t_j = -2.2 + 0.4*j, j=0..11.
__device__ __forceinline__ void bspline8(float x, float* __restrict__ o) {
    float b[11];
#pragma unroll
    for (int j = 0; j < 11; ++j) {
        float t0 = -2.2f + 0.4f * (float)j;
        float t1 = t0 + 0.4f;
        b[j] = (x >= t0 && x < t1) ? 1.0f : 0.0f;
    }
#pragma unroll
    for (int p = 1; p <= 3; ++p) {
        float inv = 1.0f / (0.4f * (float)p);   // uniform knots: denominators = 0.4*p
#pragma unroll
        for (int j = 0; j + p < 11; ++j) {
            float tj   = -2.2f + 0.4f * (float)j;
            float tjp1 = -2.2f + 0.4f * (float)(j + p + 1);
            b[j] = (x - tj) * inv * b[j] + (tjp1 - x) * inv * b[j + 1];
        }
    }
#pragma unroll
    for (int k = 0; k < 8; ++k) o[k] = b[k];
}

// h (192 x 128) -> Phi (192 x 1152): [silu(h) | spline bases, f-major x NB]
__global__ void kan_featurize(const float* __restrict__ h, float* __restrict__ Phi) {
    int idx = blockIdx.x * blockDim.x + threadIdx.x;
    if (idx >= S_DIM * H_DIM) return;
    int i = idx / H_DIM, f = idx % H_DIM;
    float x = h[idx];
    float* p = Phi + i * KPHI;
    p[f] = silu_f(x);
    float bb[8];
    bspline8(x, bb);
#pragma unroll
    for (int k = 0; k < 8; ++k) p[H_DIM + f * 8 + k] = bb[k];
}

// Weight gather in the Phi feature basis (used by pack kernels only -- cold path).
// MODE 0: pairwise KAN layer 0. wb = b0 (50 x 256), wsp = s0 (50 x 256 x 8).
//         n in [0,50)  -> A-side (feature cols 0..127); n in [50,100) -> B-side.
// MODE 1: output KAN linear. wb = l_b (128 x 128), wsp = l_s (128 x 128 x 8).
// MODE 2: dense row-major (K x N) at wb (att @ target).
template <int MODE>
__device__ __forceinline__ float loadW(const float* __restrict__ wb,
                                       const float* __restrict__ wsp,
                                       int k, int n, int N) {
    if (MODE == 2) return wb[k * N + n];
    if (MODE == 0) {
        if (n >= 100) return 0.0f;
        int d   = (n < 50) ? n : (n - 50);
        int off = (n < 50) ? 0 : 128;
        if (k < H_DIM) return wb[d * 256 + off + k];
        int kk = k - H_DIM;
        int f = kk >> 3, q = kk & 7;
        return wsp[(d * 256 + off + f) * 8 + q];
    }
    // MODE 1
    if (k < H_DIM) return wb[n * H_DIM + k];
    int kk = k - H_DIM;
    int f = kk >> 3, q = kk & 7;
    return wsp[(n * H_DIM + f) * 8 + q];
}

// Pack W (K x N, gathered) into WMMA-native interleaved layout, zero-padded to Np:
//   Wp[(k>>1) * 2*Np + n*2 + (k&1)] = W[k][n]
// so a lane's B fragment {W[k][n], W[k+1][n]} is one aligned b64 load.
template <int MODE>
__global__ void pack_weights(const float* __restrict__ wb, const float* __restrict__ wsp,
                             float* __restrict__ Wp, int K, int Np, int N) {
    int idx = blockIdx.x * blockDim.x + threadIdx.x;
    if (idx >= K * Np) return;
    int k = idx / Np, n = idx % Np;
    float v = (n < N) ? loadW<MODE>(wb, wsp, k, n, N) : 0.0f;
    Wp[(size_t)(k >> 1) * (2 * Np) + n * 2 + (k & 1)] = v;
}

// fp32 WMMA GEMM on packed weights: out(M x N) = X(M x K) @ W(K x N).
// One wave per 16x16 C tile. Branch-free inner loop:
//   4x ( b64 A-frag, b64 B-frag, v_wmma_f32_16x16x4_f32 ) per 16 k-steps.
// A layout (V_WMMA_F32_16X16X4_F32): lanes 0-15 M=0..15 K={0,1}; lanes 16-31 K={2,3}.
// B layout: lanes 0-15 N=0..15 K={0,1}; lanes 16-31 K={2,3}.
// C/D: VGPR r -> M=r (lanes 0-15) / M=r+8 (lanes 16-31), N=lane&15.
// Requires K % 16 == 0 (1152, 192 both qualify), M % 16 == 0.
template <bool RELU>
__global__ void wmma_gemm_packed(const float* __restrict__ X, const float* __restrict__ Wp,
                                 float* __restrict__ out, int K, int Np, int N) {
    int lane = threadIdx.x & 31;
    int half = lane >> 4;
    int l16  = lane & 15;
    int m0 = blockIdx.x * 16;
    int n0 = blockIdx.y * 16;
    int n  = n0 + l16;

    const float* xrow  = X + (size_t)(m0 + l16) * K + 2 * half;
    const float* wbase = Wp + (size_t)half * (2 * Np) + n * 2;

    v8f acc = {};
    for (int k0 = 0; k0 < K; k0 += 16) {
        __builtin_prefetch(xrow + k0 + 64);                       // global_prefetch_b8
        __builtin_prefetch(wbase + (size_t)(k0 + 64) * Np);
#pragma unroll
        for (int u = 0; u < 4; ++u) {
            int ks = k0 + 4 * u;
            v2f a = *(const v2f*)(xrow + ks);
            v2f b = *(const v2f*)(wbase + (size_t)ks * Np);
            acc = __builtin_amdgcn_wmma_f32_16x16x4_f32(false, a, false, b,
                                                        (short)0, acc, false, false);
        }
    }

#pragma unroll
    for (int r = 0; r < 8; ++r) {
        int row = m0 + r + half * 8;
        float v = acc[r];
        if (RELU) v = fmaxf(v, 0.0f);
        if (n < N) out[(size_t)row * N + n] = v;
    }
}

// Single-block: mf = [mean(x), mean(y), mean(target)] (3H) -> KAN^2 -> softmax -> 3 weights
__global__ void fusion_weights(const float* __restrict__ x, const float* __restrict__ y,
                               const float* __restrict__ t,
                               const float* __restrict__ fb0, const float* __restrict__ fs0,
                               const float* __restrict__ fb1, const float* __restrict__ fs1,
                               float* __restrict__ wout) {
    __shared__ float feat[384 + 384 * 8];  // [silu | bases f-major]
    __shared__ float z[64];
    __shared__ float o[3];
    int tid = threadIdx.x;
    if (tid < 384) {
        const float* src = (tid < 128) ? x : ((tid < 256) ? y : t);
        int c = tid & 127;
        float s = 0.0f;
        for (int r = 0; r < S_DIM; ++r) s += src[r * H_DIM + c];
        float v = s * (1.0f / (float)S_DIM);
        feat[tid] = silu_f(v);
        float bb[8];
        bspline8(v, bb);
#pragma unroll
        for (int k = 0; k < 8; ++k) feat[384 + tid * 8 + k] = bb[k];
    }
    __syncthreads();
    if (tid < KD) {
        float acc = 0.0f;
        for (int f = 0; f < 384; ++f) {
            acc += feat[f] * fb0[tid * 384 + f];
            const float* sp = fs0 + (tid * 384 + f) * 8;
#pragma unroll
            for (int k = 0; k < 8; ++k) acc += feat[384 + f * 8 + k] * sp[k];
        }
        z[tid] = acc;
    }
    __syncthreads();
    if (tid < 3) {
        float acc = 0.0f;
        for (int d = 0; d < KD; ++d) {
            float zz = z[d];
            acc += silu_f(zz) * fb1[tid * KD + d];
            float bb[8];
            bspline8(zz, bb);
            const float* sp = fs1 + (tid * KD + d) * 8;
#pragma unroll
            for (int k = 0; k < 8; ++k) acc += bb[k] * sp[k];
        }
        o[tid] = acc;
    }
    __syncthreads();
    if (tid == 0) {
        float mx = fmaxf(o[0], fmaxf(o[1], o[2]));
        float e0 = __expf(o[0] - mx), e1 = __expf(o[1] - mx), e2 = __expf(o[2] - mx);
        float inv = 1.0f / (e0 + e1 + e2);
        wout[0] = e0 * inv; wout[1] = e1 * inv; wout[2] = e2 * inv;
    }
}

// Per pair (i,j): z_m = A_m[i] + B_m[j] (KD dims), layer-1 spline KAN per modality,
// modality fusion + bias. AB_m row layout: [A(50) | B(50)]. 16x16 tile, LDS staged.
__global__ void pair_fusion(const float* __restrict__ ABx, const float* __restrict__ ABy,
                            const float* __restrict__ ABt,
                            const float* __restrict__ xb1, const float* __restrict__ xs1,
                            const float* __restrict__ yb1, const float* __restrict__ ys1,
                            const float* __restrict__ tb1, const float* __restrict__ ts1,
                            const float* __restrict__ wts, const float* __restrict__ bias,
                            float* __restrict__ F) {
    __shared__ float As[3][16][52];
    __shared__ float Bs[3][16][52];
    __shared__ float b1s[3][KD];
    __shared__ float s1s[3][KD * 8];
    int tj = threadIdx.x, ti = threadIdx.y;
    int tid = ti * 16 + tj;
    int i0 = blockIdx.y * 16, j0 = blockIdx.x * 16;
    const float* AB[3]  = {ABx, ABy, ABt};
    const float* b1p[3] = {xb1, yb1, tb1};
    const float* s1p[3] = {xs1, ys1, ts1};
#pragma unroll
    for (int m = 0; m < 3; ++m) {
        for (int e = tid; e < 16 * KD; e += 256) {
            int r = e / KD, d = e % KD;
            As[m][r][d] = AB[m][(i0 + r) * 100 + d];
            Bs[m][r][d] = AB[m][(j0 + r) * 100 + 50 + d];
        }
        for (int e = tid; e < KD; e += 256) b1s[m][e] = b1p[m][e];
        for (int e = tid; e < KD * 8; e += 256) s1s[m][e] = s1p[m][e];
    }
    __syncthreads();
    float fused = bias[0];
#pragma unroll
    for (int m = 0; m < 3; ++m) {
        float acc = 0.0f;
        for (int d = 0; d < KD; ++d) {
            float zz = As[m][ti][d] + Bs[m][tj][d];
            acc += silu_f(zz) * b1s[m][d];
            float bb[8];
            bspline8(zz, bb);
#pragma unroll
            for (int k = 0; k < 8; ++k) acc += bb[k] * s1s[m][d * 8 + k];
        }
        fused += wts[m] * acc;
    }
    F[(i0 + ti) * S_DIM + j0 + tj] = fused;
}

__global__ void softmax_rows(float* __restrict__ F) {
    __shared__ float sm[256];
    int row = blockIdx.x, tid = threadIdx.x;
    float v = (tid < S_DIM) ? F[row * S_DIM + tid] : -3.4e38f;
    sm[tid] = v;
    __syncthreads();
    for (int s = 128; s > 0; s >>= 1) {
        if (tid < s) sm[tid] = fmaxf(sm[tid], sm[tid + s]);
        __syncthreads();
    }
    float mx = sm[0];
    __syncthreads();
    float e = (tid < S_DIM) ? __expf(v - mx) : 0.0f;
    sm[tid] = e;
    __syncthreads();
    for (int s = 128; s > 0; s >>= 1) {
        if (tid < s) sm[tid] += sm[tid + s];
        __syncthreads();
    }
    float inv = 1.0f / sm[0];
    if (tid < S_DIM) F[row * S_DIM + tid] = e * inv;
}

extern "C" void kernel_launch(void* const* d_in, const int* in_sizes, int n_in,
                              void* d_out, int out_size, void* d_ws, size_t ws_size,
                              hipStream_t stream) {
    const float* x      = (const float*)d_in[0];
    const float* y      = (const float*)d_in[1];
    const float* target = (const float*)d_in[2];
    const float* bias   = (const float*)d_in[3];
    const float* xk_b0 = (const float*)d_in[4];
    const float* xk_s0 = (const float*)d_in[5];
    const float* xk_b1 = (const float*)d_in[6];
    const float* xk_s1 = (const float*)d_in[7];
    const float* yk_b0 = (const float*)d_in[8];
    const float* yk_s0 = (const float*)d_in[9];
    const float* yk_b1 = (const float*)d_in[10];
    const float* yk_s1 = (const float*)d_in[11];
    const float* tk_b0 = (const float*)d_in[12];
    const float* tk_s0 = (const float*)d_in[13];
    const float* tk_b1 = (const float*)d_in[14];
    const float* tk_s1 = (const float*)d_in[15];
    const float* fk_b0 = (const float*)d_in[16];
    const float* fk_s0 = (const float*)d_in[17];
    const float* fk_b1 = (const float*)d_in[18];
    const float* fk_s1 = (const float*)d_in[19];
    const float* l1_b  = (const float*)d_in[20];
    const float* l1_s  = (const float*)d_in[21];
    const float* l2_b  = (const float*)d_in[22];
    const float* l2_s  = (const float*)d_in[23];

    float* ws  = (float*)d_ws;
    float* PHI  = ws;                       // 192*1152 = 221184
    float* ABX  = PHI + S_DIM * KPHI;       // 192*100
    float* ABY  = ABX + S_DIM * 100;
    float* ABT  = ABY + S_DIM * 100;
    float* F    = ABT + S_DIM * 100;        // 192*192
    float* WTS  = F + S_DIM * S_DIM;        // 16
    float* TGT  = WTS + 16;                 // 192*128
    float* Y1   = TGT + S_DIM * H_DIM;      // 192*128
    float* WPX  = Y1 + S_DIM * H_DIM;       // 1152*112
    float* WPY  = WPX + KPHI * 112;
    float* WPT  = WPY + KPHI * 112;
    float* WPL1 = WPT + KPHI * 112;         // 1152*128
    float* WPL2 = WPL1 + KPHI * H_DIM;
    float* WPTG = WPL2 + KPHI * H_DIM;      // 192*128
    float* OUT  = (float*)d_out;

    dim3 blk32(32);
    dim3 g_ab(S_DIM / 16, 112 / 16);        // (12, 7)
    dim3 g_h(S_DIM / 16, H_DIM / 16);       // (12, 8)

    // ---- cold path: pack all weight matrices into WMMA-native layout ----
    pack_weights<0><<<(KPHI * 112 + 255) / 256, 256, 0, stream>>>(xk_b0, xk_s0, WPX, KPHI, 112, 100);
    pack_weights<0><<<(KPHI * 112 + 255) / 256, 256, 0, stream>>>(yk_b0, yk_s0, WPY, KPHI, 112, 100);
    pack_weights<0><<<(KPHI * 112 + 255) / 256, 256, 0, stream>>>(tk_b0, tk_s0, WPT, KPHI, 112, 100);
    pack_weights<1><<<(KPHI * H_DIM + 255) / 256, 256, 0, stream>>>(l1_b, l1_s, WPL1, KPHI, H_DIM, H_DIM);
    pack_weights<1><<<(KPHI * H_DIM + 255) / 256, 256, 0, stream>>>(l2_b, l2_s, WPL2, KPHI, H_DIM, H_DIM);
    pack_weights<2><<<(S_DIM * H_DIM + 255) / 256, 256, 0, stream>>>(target, nullptr, WPTG, S_DIM, H_DIM, H_DIM);

    fusion_weights<<<1, 384, 0, stream>>>(x, y, target, fk_b0, fk_s0, fk_b1, fk_s1, WTS);

    // ---- per-modality: featurize -> [A|B] = Phi @ W (fp32 WMMA) ----
    kan_featurize<<<96, 256, 0, stream>>>(x, PHI);
    wmma_gemm_packed<false><<<g_ab, blk32, 0, stream>>>(PHI, WPX, ABX, KPHI, 112, 100);
    kan_featurize<<<96, 256, 0, stream>>>(y, PHI);
    wmma_gemm_packed<false><<<g_ab, blk32, 0, stream>>>(PHI, WPY, ABY, KPHI, 112, 100);
    kan_featurize<<<96, 256, 0, stream>>>(target, PHI);
    wmma_gemm_packed<false><<<g_ab, blk32, 0, stream>>>(PHI, WPT, ABT, KPHI, 112, 100);

    // ---- pairwise layer-1 KAN + modality fusion + bias, then row softmax ----
    pair_fusion<<<dim3(S_DIM / 16, S_DIM / 16), dim3(16, 16), 0, stream>>>(
        ABX, ABY, ABT, xk_b1, xk_s1, yk_b1, yk_s1, tk_b1, tk_s1, WTS, bias, F);
    softmax_rows<<<S_DIM, 256, 0, stream>>>(F);

    // ---- tgt = att @ target (fp32 WMMA, packed dense) ----
    wmma_gemm_packed<false><<<g_h, blk32, 0, stream>>>(F, WPTG, TGT, S_DIM, H_DIM, H_DIM);

    // ---- y1 = relu(kan_linear(tgt)); y2 = relu(kan_linear(y1)) ----
    kan_featurize<<<96, 256, 0, stream>>>(TGT, PHI);
    wmma_gemm_packed<true><<<g_h, blk32, 0, stream>>>(PHI, WPL1, Y1, KPHI, H_DIM, H_DIM);
    kan_featurize<<<96, 256, 0, stream>>>(Y1, PHI);
    wmma_gemm_packed<true><<<g_h, blk32, 0, stream>>>(PHI, WPL2, OUT, KPHI, H_DIM, H_DIM);

    (void)in_sizes; (void)n_in; (void)out_size; (void)ws_size;
}